// MMD_loss_283467842756
// MI455X (gfx1250) — compile-verified
//
#include <hip/hip_runtime.h>
#include <hip/hip_bf16.h>

typedef float v2f __attribute__((ext_vector_type(2)));
typedef float v8f __attribute__((ext_vector_type(8)));

#define M_ROWS 8192
#define D_COLS 64
#define TILES_N 256          // 8192 / 32
#define TILES_M 128          // 8192 / 64
#define TILES_PER_PAIR (TILES_N * TILES_M)   // 32768
#define LDS_PAD 68           // 64 + 4 floats: keeps 16B alignment, kills bank conflicts

// ---------------------------------------------------------------------------
// Kernel 1: row squared norms for source (x2) and target (y2) into ws[0..16383]
// ---------------------------------------------------------------------------
__global__ void mmd_norms_kernel(const float* __restrict__ src,
                                 const float* __restrict__ tgt,
                                 float* __restrict__ ws) {
    int r = blockIdx.x * blockDim.x + threadIdx.x;   // 0..16383
    const float* row = (r < M_ROWS) ? (src + (size_t)r * D_COLS)
                                    : (tgt + (size_t)(r - M_ROWS) * D_COLS);
    const float4* row4 = (const float4*)row;
    float s = 0.0f;
#pragma unroll
    for (int i = 0; i < D_COLS / 4; ++i) {
        float4 v = row4[i];
        s += v.x * v.x + v.y * v.y + v.z * v.z + v.w * v.w;
    }
    ws[r] = s;
}

// ---------------------------------------------------------------------------
// Kernel 2: fused tile GEMM (f32 WMMA) + exp + tile reduction.
// blockIdx.z selects the pair: 0 = XX, 1 = YY, 2 = XY (rows=target, cols=source)
// Block = 256 threads = 8 waves; tile = 64 rows x 32 cols; wave tile = 16x16.
// ---------------------------------------------------------------------------
__global__ void mmd_tile_kernel(const float* __restrict__ src,
                                const float* __restrict__ tgt,
                                float* __restrict__ ws) {
    __shared__ float sA[64 * LDS_PAD];
    __shared__ float sB[32 * LDS_PAD];
    __shared__ float sNA[64];
    __shared__ float sNB[32];
    __shared__ float sRed[256];

    const float* x2 = ws;            // source norms
    const float* y2 = ws + M_ROWS;   // target norms

    const float* A; const float* B; const float* nA; const float* nB;
    int p = blockIdx.z;
    if (p == 0)      { A = src; B = src; nA = x2; nB = x2; }
    else if (p == 1) { A = tgt; B = tgt; nA = y2; nB = y2; }
    else             { A = tgt; B = src; nA = y2; nB = x2; }

    const int t = threadIdx.x;
    const int rowBaseA = blockIdx.y * 64;
    const int rowBaseB = blockIdx.x * 32;

    // Load A tile: 64x64 f32 = 1024 float4, 4 per thread.
#pragma unroll
    for (int i = 0; i < 4; ++i) {
        int idx = t + i * 256;
        int r = idx >> 4;            // 0..63
        int c = (idx & 15) << 2;     // 0..60
        *(float4*)&sA[r * LDS_PAD + c] =
            *(const float4*)&A[(size_t)(rowBaseA + r) * D_COLS + c];
    }
    // Load B tile: 32x64 f32 = 512 float4, 2 per thread.
#pragma unroll
    for (int i = 0; i < 2; ++i) {
        int idx = t + i * 256;
        int r = idx >> 4;            // 0..31
        int c = (idx & 15) << 2;
        *(float4*)&sB[r * LDS_PAD + c] =
            *(const float4*)&B[(size_t)(rowBaseB + r) * D_COLS + c];
    }
    // Row norms for the tile.
    if (t < 64)               sNA[t]      = nA[rowBaseA + t];
    else if (t < 96)          sNB[t - 64] = nB[rowBaseB + (t - 64)];
    __syncthreads();

    const int lane   = t & 31;
    const int wave   = t >> 5;       // 0..7
    const int wm     = wave & 3;     // row wave-tile  (4 x 16 = 64 rows)
    const int wn     = wave >> 2;    // col wave-tile  (2 x 16 = 32 cols)
    const int laneLo = lane & 15;
    const int laneHi = lane >> 4;

    // A 16x4 f32 operand: lanes 0-15 hold K={0,1} for M=lane; lanes 16-31 K={2,3}.
    // B 4x16 f32 operand mirrors it with N=laneLo (B[k][n] = rowB_n[k]).
    const float* arow = &sA[(wm * 16 + laneLo) * LDS_PAD];
    const float* brow = &sB[(wn * 16 + laneLo) * LDS_PAD];

    v8f c = {};
#pragma unroll
    for (int k0 = 0; k0 < D_COLS; k0 += 4) {
        int ka = k0 + 2 * laneHi;
        v2f a; a.x = arow[ka]; a.y = arow[ka + 1];
        v2f b; b.x = brow[ka]; b.y = brow[ka + 1];
        c = __builtin_amdgcn_wmma_f32_16x16x4_f32(
                /*neg_a=*/false, a, /*neg_b=*/false, b,
                /*c_mod=*/(short)0, c, /*reuse_a=*/false, /*reuse_b=*/false);
    }

    // D[m][n]: VGPR v holds m_local = v + 8*laneHi, n_local = laneLo.
    const float cn = sNB[wn * 16 + laneLo];
    float acc = 0.0f;
#pragma unroll
    for (int v = 0; v < 8; ++v) {
        int mloc = v + 8 * laneHi;
        float d = sNA[wm * 16 + mloc] + cn - 2.0f * c[v];
        d = d > 0.0f ? d : 0.0f;              // jnp.maximum(d, 0)
        acc += __expf(-0.5f * d);             // alpha = 1/(2*FIX_SIGMA) = 0.5
    }

    // Deterministic block reduction.
    sRed[t] = acc;
    __syncthreads();
#pragma unroll
    for (int s = 128; s > 0; s >>= 1) {
        if (t < s) sRed[t] += sRed[t + s];
        __syncthreads();
    }
    if (t == 0) {
        int tileId = blockIdx.y * gridDim.x + blockIdx.x;
        ws[2 * M_ROWS + (size_t)p * TILES_PER_PAIR + tileId] = sRed[0];
    }
}

// ---------------------------------------------------------------------------
// Kernel 3: deterministic final reduction + MMD combine.
// ---------------------------------------------------------------------------
__global__ void mmd_final_kernel(const float* __restrict__ ws,
                                 float* __restrict__ out) {
    __shared__ double sRed[256];
    const float* partial = ws + 2 * M_ROWS;
    const int t = threadIdx.x;
    double res[3];
#pragma unroll
    for (int p = 0; p < 3; ++p) {
        double s = 0.0;
        for (int i = t; i < TILES_PER_PAIR; i += 256)
            s += (double)partial[(size_t)p * TILES_PER_PAIR + i];
        sRed[t] = s;
        __syncthreads();
        for (int k = 128; k > 0; k >>= 1) {
            if (t < k) sRed[t] += sRed[t + k];
            __syncthreads();
        }
        if (t == 0) res[p] = sRed[0];
        __syncthreads();
    }
    if (t == 0) {
        const double m = (double)M_ROWS;
        // Diagonals of XX/YY are exp(-0.5*0) == 1 -> subtract m analytically.
        double xx = (res[0] - m) / (m * (m - 1.0));
        double yy = (res[1] - m) / (m * (m - 1.0));
        double xy = res[2] / (m * m);
        out[0] = (float)(xx + yy - 2.0 * xy);
    }
}

extern "C" void kernel_launch(void* const* d_in, const int* in_sizes, int n_in,
                              void* d_out, int out_size, void* d_ws, size_t ws_size,
                              hipStream_t stream) {
    const float* src = (const float*)d_in[0];   // source (8192 x 64) f32
    const float* tgt = (const float*)d_in[1];   // target (8192 x 64) f32
    float* ws  = (float*)d_ws;                  // [x2 | y2 | partials(3*32768)]
    float* out = (float*)d_out;

    mmd_norms_kernel<<<(2 * M_ROWS) / 256, 256, 0, stream>>>(src, tgt, ws);

    dim3 grid(TILES_N, TILES_M, 3);
    mmd_tile_kernel<<<grid, 256, 0, stream>>>(src, tgt, ws);

    mmd_final_kernel<<<1, 256, 0, stream>>>(ws, out);
}